// MutiscaleGCN_16982300688883
// MI455X (gfx1250) — compile-verified
//
#include <hip/hip_runtime.h>
#include <stdint.h>

#define N_NODES 4096
#define N_EDGES 131072
#define IN_CH   512
#define HID     512
#define HC      1536          // 3*HID concat width
#define NWORDS  128           // 4096 bits / 32

typedef __attribute__((ext_vector_type(2))) float v2f;
typedef __attribute__((ext_vector_type(8))) float v8f;

// ---------------------------------------------------------------------------
// Build adjacency bitmap + (multiplicity-aware) in-degree at destination.
// ---------------------------------------------------------------------------
__global__ void gcn_build_adj(const int* __restrict__ ei,
                              unsigned* __restrict__ abit,
                              int* __restrict__ degE) {
    int e = blockIdx.x * blockDim.x + threadIdx.x;
    if (e >= N_EDGES) return;
    int r = ei[e];
    int c = ei[N_EDGES + e];
    atomicOr(&abit[r * NWORDS + (c >> 5)], 1u << (c & 31));
    atomicAdd(&degE[c], 1);
}

__global__ void gcn_dinv_edge(const int* __restrict__ degE,
                              float* __restrict__ dinv) {
    int i = blockIdx.x * blockDim.x + threadIdx.x;
    if (i < N_NODES) dinv[i] = rsqrtf((float)(degE[i] + 1));   // +1 self loop
}

// ---------------------------------------------------------------------------
// Boolean matmul on bitset rows: O[i,:] = OR_{k : L[i,k]} R[k,:]
// One 128-thread block per row; thread t owns output word t.
// ---------------------------------------------------------------------------
__global__ void __launch_bounds__(128)
gcn_bool_mm(const unsigned* __restrict__ L, const unsigned* __restrict__ R,
            unsigned* __restrict__ O) {
    __shared__ unsigned rowL[NWORDS];
    const int i = blockIdx.x;
    const int t = threadIdx.x;
    rowL[t] = L[i * NWORDS + t];
    __syncthreads();
    unsigned acc = 0u;
    for (int w = 0; w < NWORDS; ++w) {
        unsigned bits = rowL[w];
        while (bits) {
            int b = __builtin_ctz(bits);
            bits &= bits - 1u;
            int k = (w << 5) + b;
            acc |= R[k * NWORDS + t];          // coalesced 512B gather per k
        }
    }
    O[i * NWORDS + t] = acc;
}

// Column sums of binary matrix (+1 self loop) -> dinv
__global__ void gcn_col_dinv(const unsigned* __restrict__ B,
                             float* __restrict__ dinv) {
    int j = blockIdx.x * blockDim.x + threadIdx.x;
    if (j >= N_NODES) return;
    const int w = j >> 5;
    const unsigned m = 1u << (j & 31);
    int cnt = 1;                               // identity term
    for (int i = 0; i < N_NODES; ++i)
        cnt += (B[i * NWORDS + w] & m) ? 1 : 0;
    dinv[j] = rsqrtf((float)cnt);
}

// ---------------------------------------------------------------------------
// Dense f32 GEMM: each wave computes a 16x64 strip (4 accumulators) so the
// A fragment is reused across 4 V_WMMA_F32_16X16X4_F32 issues.
//   C[m, colOff+n] = rowscale[m] * (A@B)[m,n] + bias[n]
// ---------------------------------------------------------------------------
__global__ void __launch_bounds__(128)
gcn_gemm_xw(const float* __restrict__ A, const float* __restrict__ B,
            float* __restrict__ C, int M, int N, int K, int ldc, int colOff,
            const float* __restrict__ bias, const float* __restrict__ rowscale) {
    const int lane   = threadIdx.x & 31;
    const int wave   = threadIdx.x >> 5;
    const int tilesN = N >> 6;                 // strips of 64 columns
    const int strip  = blockIdx.x * 4 + wave;
    const int tm = strip / tilesN;
    const int tn = strip % tilesN;
    const int row  = tm * 16 + (lane & 15);
    const int col0 = tn * 64 + (lane & 15);
    const int kh   = (lane >> 4) << 1;         // lane-half selects K pair {0,1} vs {2,3}
    const float* __restrict__ Arow = A + (size_t)row * K;

    v8f acc0 = {}, acc1 = {}, acc2 = {}, acc3 = {};
    for (int k = 0; k < K; k += 4) {
        v2f a;
        a.x = Arow[k + kh];
        a.y = Arow[k + kh + 1];
        const float* __restrict__ B0 = B + (size_t)(k + kh) * N + col0;
        const float* __restrict__ B1 = B0 + N;
        v2f b0, b1, b2, b3;
        b0.x = B0[0];  b0.y = B1[0];
        b1.x = B0[16]; b1.y = B1[16];
        b2.x = B0[32]; b2.y = B1[32];
        b3.x = B0[48]; b3.y = B1[48];
        acc0 = __builtin_amdgcn_wmma_f32_16x16x4_f32(false, a, false, b0, (short)0, acc0, false, false);
        acc1 = __builtin_amdgcn_wmma_f32_16x16x4_f32(false, a, false, b1, (short)0, acc1, false, false);
        acc2 = __builtin_amdgcn_wmma_f32_16x16x4_f32(false, a, false, b2, (short)0, acc2, false, false);
        acc3 = __builtin_amdgcn_wmma_f32_16x16x4_f32(false, a, false, b3, (short)0, acc3, false, false);
    }
    const int m0 = tm * 16 + ((lane >> 4) << 3);
    v8f accs[4] = {acc0, acc1, acc2, acc3};
    #pragma unroll
    for (int t = 0; t < 4; ++t) {
        const int col = col0 + 16 * t;
        const float bc = bias ? bias[col] : 0.0f;
        #pragma unroll
        for (int v = 0; v < 8; ++v) {
            const int m = m0 + v;
            float val = accs[t][v];
            if (rowscale) val *= rowscale[m];
            C[(size_t)m * ldc + colOff + col] = val + bc;
        }
    }
}

// ---------------------------------------------------------------------------
// Dense-support GCN aggregation:  H[j,:] = relu(dinv[j]*((B+I)^T @ Xs)[j,:] + b)
// A-fragment synthesized from the bitmap: value = bit(B[i,j]) + (i==j).
// The bitmap word index jw = tm>>1 is wave-uniform: load the 4 K-slice words
// once per step via a readfirstlane-pinned pointer, select per lane half.
// ---------------------------------------------------------------------------
__global__ void __launch_bounds__(128)
gcn_spmm_bits(const unsigned* __restrict__ Bbit, const float* __restrict__ Xs,
              float* __restrict__ H, const float* __restrict__ dinv,
              const float* __restrict__ bias, int colOff) {
    const int lane   = threadIdx.x & 31;
    const int wave   = threadIdx.x >> 5;
    const int tilesN = HID >> 6;               // 8 strips of 64 columns
    const int strip  = blockIdx.x * 4 + wave;
    const int tm = strip / tilesN;
    const int tn = strip % tilesN;
    const int j    = tm * 16 + (lane & 15);    // output row (column of B)
    const int col0 = tn * 64 + (lane & 15);
    const int kh   = (lane >> 4) << 1;
    const int jw   = __builtin_amdgcn_readfirstlane(tm >> 1);  // wave-uniform word
    const int js   = j & 31;
    const unsigned* __restrict__ wp = Bbit + jw;

    v8f acc0 = {}, acc1 = {}, acc2 = {}, acc3 = {};
    for (int k = 0; k < N_NODES; k += 4) {
        // 4 wave-uniform bitmap words for K-slice [k, k+4)
        const unsigned w0 = wp[(size_t)(k + 0) * NWORDS];
        const unsigned w1 = wp[(size_t)(k + 1) * NWORDS];
        const unsigned w2 = wp[(size_t)(k + 2) * NWORDS];
        const unsigned w3 = wp[(size_t)(k + 3) * NWORDS];
        const int i0 = k + kh;
        const int i1 = i0 + 1;
        const unsigned wa = (kh == 0) ? w0 : w2;
        const unsigned wb = (kh == 0) ? w1 : w3;
        v2f a;
        a.x = (float)((wa >> js) & 1u) + (i0 == j ? 1.0f : 0.0f);
        a.y = (float)((wb >> js) & 1u) + (i1 == j ? 1.0f : 0.0f);
        const float* __restrict__ X0 = Xs + (size_t)i0 * HID + col0;
        const float* __restrict__ X1 = X0 + HID;
        v2f b0, b1, b2, b3;
        b0.x = X0[0];  b0.y = X1[0];
        b1.x = X0[16]; b1.y = X1[16];
        b2.x = X0[32]; b2.y = X1[32];
        b3.x = X0[48]; b3.y = X1[48];
        acc0 = __builtin_amdgcn_wmma_f32_16x16x4_f32(false, a, false, b0, (short)0, acc0, false, false);
        acc1 = __builtin_amdgcn_wmma_f32_16x16x4_f32(false, a, false, b1, (short)0, acc1, false, false);
        acc2 = __builtin_amdgcn_wmma_f32_16x16x4_f32(false, a, false, b2, (short)0, acc2, false, false);
        acc3 = __builtin_amdgcn_wmma_f32_16x16x4_f32(false, a, false, b3, (short)0, acc3, false, false);
    }
    const int m0 = tm * 16 + ((lane >> 4) << 3);
    v8f accs[4] = {acc0, acc1, acc2, acc3};
    #pragma unroll
    for (int t = 0; t < 4; ++t) {
        const int col = col0 + 16 * t;
        const float bc = bias[col];
        #pragma unroll
        for (int v = 0; v < 8; ++v) {
            const int m = m0 + v;
            float val = dinv[m] * accs[t][v] + bc;
            H[(size_t)m * HC + colOff + col] = fmaxf(val, 0.0f);
        }
    }
}

// ---------------------------------------------------------------------------
// Edge-list GCN (keeps duplicate-edge multiplicity, matching reference).
// ---------------------------------------------------------------------------
__global__ void gcn_init_agg(const float* __restrict__ xw1,
                             const float* __restrict__ dinvE,
                             float* __restrict__ agg) {
    int t = blockIdx.x * blockDim.x + threadIdx.x;     // 4096*512
    int n = t >> 9;
    float d = dinvE[n];
    agg[t] = d * d * xw1[t];                           // self-loop contribution
}

__global__ void __launch_bounds__(128)
gcn_edge_scatter(const int* __restrict__ ei, const float* __restrict__ xw1,
                 const float* __restrict__ dinvE, float* __restrict__ agg) {
    const int e = blockIdx.x;
    const int r = ei[e];
    const int c = ei[N_EDGES + e];
    const float s = dinvE[r] * dinvE[c];
    const float* __restrict__ src = xw1 + (size_t)r * HID;
    float* __restrict__ dst = agg + (size_t)c * HID;
    for (int ch = threadIdx.x; ch < HID; ch += 128)
        atomicAdd(&dst[ch], s * src[ch]);
}

__global__ void gcn_h1_write(const float* __restrict__ agg,
                             const float* __restrict__ b1,
                             float* __restrict__ H) {
    int t = blockIdx.x * blockDim.x + threadIdx.x;     // 4096*512
    int n = t >> 9;
    int c = t & (HID - 1);
    H[(size_t)n * HC + c] = fmaxf(agg[t] + b1[c], 0.0f);
}

// ---------------------------------------------------------------------------
// Fusion GEMM: out = fw*relu(concat@Wf + bf) + (1-fw)*original
// Same 16x64-per-wave strip tiling.
// ---------------------------------------------------------------------------
__global__ void __launch_bounds__(128)
gcn_gemm_fusion(const float* __restrict__ A, const float* __restrict__ B,
                const float* __restrict__ bias, const float* __restrict__ orig,
                const float* __restrict__ fwp, float* __restrict__ out) {
    const int lane   = threadIdx.x & 31;
    const int wave   = threadIdx.x >> 5;
    const int tilesN = HID >> 6;
    const int strip  = blockIdx.x * 4 + wave;
    const int tm = strip / tilesN;
    const int tn = strip % tilesN;
    const int row  = tm * 16 + (lane & 15);
    const int col0 = tn * 64 + (lane & 15);
    const int kh   = (lane >> 4) << 1;
    const float* __restrict__ Arow = A + (size_t)row * HC;

    v8f acc0 = {}, acc1 = {}, acc2 = {}, acc3 = {};
    for (int k = 0; k < HC; k += 4) {
        v2f a;
        a.x = Arow[k + kh];
        a.y = Arow[k + kh + 1];
        const float* __restrict__ B0 = B + (size_t)(k + kh) * HID + col0;
        const float* __restrict__ B1 = B0 + HID;
        v2f b0, b1, b2, b3;
        b0.x = B0[0];  b0.y = B1[0];
        b1.x = B0[16]; b1.y = B1[16];
        b2.x = B0[32]; b2.y = B1[32];
        b3.x = B0[48]; b3.y = B1[48];
        acc0 = __builtin_amdgcn_wmma_f32_16x16x4_f32(false, a, false, b0, (short)0, acc0, false, false);
        acc1 = __builtin_amdgcn_wmma_f32_16x16x4_f32(false, a, false, b1, (short)0, acc1, false, false);
        acc2 = __builtin_amdgcn_wmma_f32_16x16x4_f32(false, a, false, b2, (short)0, acc2, false, false);
        acc3 = __builtin_amdgcn_wmma_f32_16x16x4_f32(false, a, false, b3, (short)0, acc3, false, false);
    }
    const float fw = *fwp;
    const int m0 = tm * 16 + ((lane >> 4) << 3);
    v8f accs[4] = {acc0, acc1, acc2, acc3};
    #pragma unroll
    for (int t = 0; t < 4; ++t) {
        const int col = col0 + 16 * t;
        const float bc = bias[col];
        #pragma unroll
        for (int v = 0; v < 8; ++v) {
            const int m = m0 + v;
            float h = fmaxf(accs[t][v] + bc, 0.0f);
            out[(size_t)m * HID + col] = fw * h + (1.0f - fw) * orig[(size_t)m * HID + col];
        }
    }
}

// ---------------------------------------------------------------------------
extern "C" void kernel_launch(void* const* d_in, const int* in_sizes, int n_in,
                              void* d_out, int out_size, void* d_ws, size_t ws_size,
                              hipStream_t stream) {
    (void)in_sizes; (void)n_in; (void)out_size; (void)ws_size;
    const float* x  = (const float*)d_in[0];
    const int*   ei = (const int*)  d_in[1];
    const float* Wp = (const float*)d_in[2];
    const float* bp = (const float*)d_in[3];
    const float* W1 = (const float*)d_in[4];
    const float* b1 = (const float*)d_in[5];
    const float* W2 = (const float*)d_in[6];
    const float* b2 = (const float*)d_in[7];
    const float* W3 = (const float*)d_in[8];
    const float* b3 = (const float*)d_in[9];
    const float* Wf = (const float*)d_in[10];
    const float* bf = (const float*)d_in[11];
    const float* fw = (const float*)d_in[12];

    char* ws = (char*)d_ws;
    const size_t SZ_BIT = (size_t)N_NODES * NWORDS * 4;     // 2 MB
    const size_t SZ_VEC = (size_t)N_NODES * 4;              // 16 KB
    const size_t SZ_MAT = (size_t)N_NODES * HID * 4;        // 8 MB

    size_t off = 0;
    unsigned* Abit  = (unsigned*)(ws + off); off += SZ_BIT;
    int*      degE  = (int*)     (ws + off); off += SZ_VEC;
    unsigned* B2b   = (unsigned*)(ws + off); off += SZ_BIT;
    unsigned* B3b   = (unsigned*)(ws + off); off += SZ_BIT;
    float*    dinvE = (float*)   (ws + off); off += SZ_VEC;
    float*    dinv2 = (float*)   (ws + off); off += SZ_VEC;
    float*    dinv3 = (float*)   (ws + off); off += SZ_VEC;
    float*    orig  = (float*)   (ws + off); off += SZ_MAT;
    float*    xw1   = (float*)   (ws + off); off += SZ_MAT;
    float*    xw2   = (float*)   (ws + off); off += SZ_MAT;
    float*    xw3   = (float*)   (ws + off); off += SZ_MAT;
    float*    agg   = (float*)   (ws + off); off += SZ_MAT;
    float*    hcon  = (float*)   (ws + off); off += (size_t)N_NODES * HC * 4;

    // Zero adjacency bitmap + edge degree (contiguous region).
    hipMemsetAsync(ws, 0, SZ_BIT + SZ_VEC, stream);

    // Graph structure.
    gcn_build_adj<<<N_EDGES / 256, 256, 0, stream>>>(ei, Abit, degE);
    gcn_dinv_edge<<<N_NODES / 256, 256, 0, stream>>>(degE, dinvE);
    gcn_bool_mm<<<N_NODES, 128, 0, stream>>>(Abit, Abit, B2b);   // 2-hop support
    gcn_bool_mm<<<N_NODES, 128, 0, stream>>>(B2b,  Abit, B3b);   // 3-hop support
    gcn_col_dinv<<<N_NODES / 128, 128, 0, stream>>>(B2b, dinv2);
    gcn_col_dinv<<<N_NODES / 128, 128, 0, stream>>>(B3b, dinv3);

    // Feature projections (f32 WMMA). (4096/16)*(512/64) strips / 4 waves.
    const int GB = (N_NODES / 16) * (HID / 64) / 4;              // 512 blocks
    gcn_gemm_xw<<<GB, 128, 0, stream>>>(x, Wp, orig, N_NODES, HID, IN_CH, HID, 0, bp, nullptr);
    gcn_gemm_xw<<<GB, 128, 0, stream>>>(x, W1, xw1,  N_NODES, HID, IN_CH, HID, 0, nullptr, nullptr);
    gcn_gemm_xw<<<GB, 128, 0, stream>>>(x, W2, xw2,  N_NODES, HID, IN_CH, HID, 0, nullptr, dinv2);
    gcn_gemm_xw<<<GB, 128, 0, stream>>>(x, W3, xw3,  N_NODES, HID, IN_CH, HID, 0, nullptr, dinv3);

    // Edge-list GCN branch (h1) -> concat columns [0, 512).
    gcn_init_agg<<<(N_NODES * HID) / 256, 256, 0, stream>>>(xw1, dinvE, agg);
    gcn_edge_scatter<<<N_EDGES, 128, 0, stream>>>(ei, xw1, dinvE, agg);
    gcn_h1_write<<<(N_NODES * HID) / 256, 256, 0, stream>>>(agg, b1, hcon);

    // Dense-support GCN branches (h2, h3) -> concat columns [512,1024),[1024,1536).
    gcn_spmm_bits<<<GB, 128, 0, stream>>>(B2b, xw2, hcon, dinv2, b2, 512);
    gcn_spmm_bits<<<GB, 128, 0, stream>>>(B3b, xw3, hcon, dinv3, b3, 1024);

    // Fusion GEMM + residual mix -> d_out.
    gcn_gemm_fusion<<<GB, 128, 0, stream>>>(hcon, Wf, bf, orig, fw, (float*)d_out);
}